// LightGCN_83708912599774
// MI455X (gfx1250) — compile-verified
//
#include <hip/hip_runtime.h>
#include <hip/hip_bf16.h>
#include <stdint.h>

// ---------------- problem constants (match reference) ----------------
#define N_USERS 100000
#define N_ITEMS 50000
#define N_NODES (N_USERS + N_ITEMS)
#define N_EDGESC 10000000
#define DIM 64
#define EPB 1024   // edges staged per block (12 KB LDS)

// ---------------- CDNA5 async global->LDS path (guarded) ----------------
#if defined(__gfx1250__) && __has_builtin(__builtin_amdgcn_global_load_async_to_lds_b128) && \
    __has_builtin(__builtin_amdgcn_s_wait_asynccnt)
#define USE_ASYNC_LDS 1
#else
#define USE_ASYNC_LDS 0
#endif

#if USE_ASYNC_LDS
// Builtin signature (from clang diagnostic): param0 is `int __vector(4) __device__ *`
// i.e. non-const v4i* in the global address space; param1 is the LDS-side pointer.
typedef int v4i __attribute__((ext_vector_type(4)));
typedef __attribute__((address_space(1))) v4i gv4i_t;   // global
typedef __attribute__((address_space(3))) v4i lv4i_t;   // LDS

// cpol: gfx12 encoding, TH in bits[2:0]; TH=1 => non-temporal. Edge arrays are
// streamed once per SpMM pass (120 MB/layer) — keep them from evicting the
// L2-resident feature buffers (~115 MB of a 192 MB L2).
#define EDGE_CPOL 1

__device__ __forceinline__ void async_copy16(const void* g, void* l) {
  // AS(1) pointer value == 64-bit global VA; generic LDS address low 32 bits
  // are the raw LDS byte offset (ISA 10.2), so integer truncation is correct.
  gv4i_t* gp = (gv4i_t*)(uintptr_t)g;
  lv4i_t* lp = (lv4i_t*)(uint32_t)(uintptr_t)l;
  __builtin_amdgcn_global_load_async_to_lds_b128(gp, lp, 0, EDGE_CPOL);
}
#endif

// ---------------- kernels ----------------

// out[i] = x[i] = concat(user_emb, item_emb)   (float4-vectorized; counts divisible by 4)
__global__ void __launch_bounds__(256)
init_concat(const float4* __restrict__ u, const float4* __restrict__ it,
            float4* __restrict__ x, float4* __restrict__ out, int nU4, int nTot4) {
  int i = blockIdx.x * blockDim.x + threadIdx.x;
  if (i < nTot4) {
    float4 v = (i < nU4) ? u[i] : it[i - nU4];
    x[i]   = v;
    out[i] = v;
  }
}

__global__ void __launch_bounds__(256)
zero_kernel(float4* __restrict__ p, int n4) {
  int i = blockIdx.x * blockDim.x + threadIdx.x;
  if (i < n4) {
    float4 z; z.x = 0.f; z.y = 0.f; z.z = 0.f; z.w = 0.f;
    p[i] = z;
  }
}

// y[row] += val * x[col]; one wave per edge, lane handles 2 consecutive floats.
__global__ void __launch_bounds__(256)
spmm_kernel(const int* __restrict__ rows, const int* __restrict__ cols,
            const float* __restrict__ vals, const float* __restrict__ x,
            float* __restrict__ y, int nEdges) {
  __shared__ __align__(16) int   sRows[EPB];
  __shared__ __align__(16) int   sCols[EPB];
  __shared__ __align__(16) float sVals[EPB];

  const int base = blockIdx.x * EPB;
  const int tid  = threadIdx.x;
  int nChunk = nEdges - base;
  if (nChunk > EPB) nChunk = EPB;     // N_EDGES % 4 == 0, so every chunk is a multiple of 4

  // ---- stage edge data into LDS (async on CDNA5, ds_store_b128 fallback) ----
  const int t4 = tid * 4;
  if (t4 < nChunk) {
#if USE_ASYNC_LDS
    async_copy16(rows + base + t4, &sRows[t4]);
    async_copy16(cols + base + t4, &sCols[t4]);
    async_copy16(vals + base + t4, &sVals[t4]);
#else
    *(int4*)&sRows[t4]   = *(const int4*)(rows + base + t4);
    *(int4*)&sCols[t4]   = *(const int4*)(cols + base + t4);
    *(float4*)&sVals[t4] = *(const float4*)(vals + base + t4);
#endif
  }
#if USE_ASYNC_LDS
  __builtin_amdgcn_s_wait_asynccnt(0);
#endif
  __syncthreads();

  // ---- process: 8 waves/block, 128 edges/wave, 32 lanes x float2 = 64-wide row ----
  const int wave = tid >> 5;
  const int lane = tid & 31;
  int e    = wave * (EPB / 8);
  int eEnd = e + (EPB / 8);
  if (eEnd > nChunk) eEnd = nChunk;

#pragma unroll 2
  for (; e < eEnd; ++e) {
    // prefetch the gather row a few edges ahead (global_prefetch_b8);
    // locality=3: gathered x-rows are reused ~64x across the edge set
    if (e + 8 < eEnd) {
      __builtin_prefetch(x + (size_t)sCols[e + 8] * DIM + lane * 2, 0, 3);
    }
    const int   c = sCols[e];     // LDS broadcast reads (conflict-free)
    const int   r = sRows[e];
    const float v = sVals[e];
    const float2 xv = *(const float2*)(x + (size_t)c * DIM + lane * 2);
    float* yp = y + (size_t)r * DIM + lane * 2;
    unsafeAtomicAdd(yp,     v * xv.x);   // global_atomic_add_f32, no return
    unsafeAtomicAdd(yp + 1, v * xv.y);
  }
}

__global__ void __launch_bounds__(256)
accum_kernel(float4* __restrict__ out, const float4* __restrict__ y, int n4) {
  int i = blockIdx.x * blockDim.x + threadIdx.x;
  if (i < n4) {
    float4 o = out[i]; float4 a = y[i];
    o.x += a.x; o.y += a.y; o.z += a.z; o.w += a.w;
    out[i] = o;
  }
}

__global__ void __launch_bounds__(256)
accum_scale_kernel(float4* __restrict__ out, const float4* __restrict__ y, int n4) {
  int i = blockIdx.x * blockDim.x + threadIdx.x;
  if (i < n4) {
    float4 o = out[i]; float4 a = y[i];
    o.x = (o.x + a.x) * 0.25f; o.y = (o.y + a.y) * 0.25f;
    o.z = (o.z + a.z) * 0.25f; o.w = (o.w + a.w) * 0.25f;
    out[i] = o;
  }
}

// ---------------- host side ----------------
extern "C" void kernel_launch(void* const* d_in, const int* in_sizes, int n_in,
                              void* d_out, int out_size, void* d_ws, size_t ws_size,
                              hipStream_t stream) {
  (void)in_sizes; (void)n_in; (void)out_size; (void)ws_size;

  const float* user_emb  = (const float*)d_in[0];
  const float* item_emb  = (const float*)d_in[1];
  const float* edge_vals = (const float*)d_in[2];
  const int*   edge_rows = (const int*)d_in[3];
  const int*   edge_cols = (const int*)d_in[4];
  float* out = (float*)d_out;   // light_out flat == users||items, same layout as acc

  // workspace: two ping-pong feature buffers (2 * 38.4 MB); both L2-resident
  float* bufA = (float*)d_ws;
  float* bufB = bufA + (size_t)N_NODES * DIM;

  const int nTot4 = N_NODES * DIM / 4;     // 2.4M float4
  const int nU4   = N_USERS * DIM / 4;
  const int eBlk  = 256;
  const int eGrid = (nTot4 + eBlk - 1) / eBlk;
  const int nChunks = (N_EDGESC + EPB - 1) / EPB;

  init_concat<<<eGrid, eBlk, 0, stream>>>((const float4*)user_emb, (const float4*)item_emb,
                                          (float4*)bufA, (float4*)out, nU4, nTot4);

  float* xin  = bufA;
  float* yout = bufB;
  for (int layer = 0; layer < 3; ++layer) {
    zero_kernel<<<eGrid, eBlk, 0, stream>>>((float4*)yout, nTot4);
    spmm_kernel<<<nChunks, 256, 0, stream>>>(edge_rows, edge_cols, edge_vals,
                                             xin, yout, N_EDGESC);
    if (layer < 2)
      accum_kernel<<<eGrid, eBlk, 0, stream>>>((float4*)out, (const float4*)yout, nTot4);
    else
      accum_scale_kernel<<<eGrid, eBlk, 0, stream>>>((float4*)out, (const float4*)yout, nTot4);
    float* t = xin; xin = yout; yout = t;   // ping-pong features
  }
}